// AbstractiveTextSummarize_20040317403604
// MI455X (gfx1250) — compile-verified
//
#include <hip/hip_runtime.h>
#include <math.h>

typedef float v2f __attribute__((ext_vector_type(2)));
typedef float v8f __attribute__((ext_vector_type(8)));

#define VCAB 32000
#define VEXTN 32100
#define TOTF 224
#define STN 32

__device__ __forceinline__ float sigmf(float x) { return 1.0f / (1.0f + __expf(-x)); }

// ---------------------------------------------------------------------------
// Generic fp32 WMMA GEMM: C[M,N] = act(A[M,K] * W[N,K]^T + bias[N])
// One wave per 16x16 tile. act: 0=none, 1=tanh.
// Out-of-range rows/cols are handled by POINTER CLAMPING only: a clamped A row
// corrupts only D rows >= M and a clamped B col corrupts only D cols >= N,
// both of which are write-guarded. Loads are therefore unconditional b64s.
// ---------------------------------------------------------------------------
__global__ __launch_bounds__(32)
void gemm_f32_wmma(const float* __restrict__ A, int lda, int M,
                   const float* __restrict__ W, int ldw, int N, int K,
                   const float* __restrict__ bias, float* __restrict__ C,
                   int ldc, int act)
{
  const int lane = threadIdx.x & 31;
  const int m0 = blockIdx.x * 16;
  const int n0 = blockIdx.y * 16;
  const int hf = lane >> 4;          // K-pair selector (K=0,1 vs K=2,3)
  const int sub = lane & 15;
  const int mr_c = min(m0 + sub, M - 1);
  const int nr_c = min(n0 + sub, N - 1);
  const float* arow = A + (size_t)mr_c * lda + hf * 2;
  const float* wrow = W + (size_t)nr_c * ldw + hf * 2;
  v8f acc = {};
#pragma unroll 4
  for (int k = 0; k < K; k += 4) {
    v2f a = *(const v2f*)(arow + k);
    v2f b = *(const v2f*)(wrow + k);
    acc = __builtin_amdgcn_wmma_f32_16x16x4_f32(false, a, false, b, (short)0,
                                                acc, false, false);
  }
  const int nr = n0 + sub;
  if (nr >= N) return;
  const float bv = bias ? bias[nr] : 0.f;
#pragma unroll
  for (int v = 0; v < 8; ++v) {
    const int m = m0 + v + hf * 8;
    if (m < M) {
      float x = acc[v] + bv;
      if (act == 1) x = tanhf(x);
      C[(size_t)m * ldc + nr] = x;
    }
  }
}

// ---------------------------------------------------------------------------
// Feature-rich embedding concat -> emb_flat[4096, 224]
// ---------------------------------------------------------------------------
__global__ void embed_concat(const int* __restrict__ iid, const int* __restrict__ pid,
                             const int* __restrict__ nid, const int* __restrict__ tfid,
                             const float* __restrict__ wemb, const float* __restrict__ pemb,
                             const float* __restrict__ nemb, const float* __restrict__ temb,
                             float* __restrict__ outp)
{
  int idx = blockIdx.x * blockDim.x + threadIdx.x;
  if (idx >= 4096 * TOTF) return;
  int row = idx / TOTF, c = idx - row * TOTF;
  float v;
  if (c < 128) {
    int tok = iid[row]; tok = (tok >= VCAB) ? 2 : (tok < 0 ? 0 : tok);
    v = wemb[(size_t)tok * 128 + c];
  } else if (c < 160) {
    int p = pid[row]; p = p < 0 ? 0 : (p > 31 ? 31 : p);
    v = pemb[p * 32 + (c - 128)];
  } else if (c < 192) {
    int p = nid[row]; p = p < 0 ? 0 : (p > 15 ? 15 : p);
    v = nemb[p * 32 + (c - 160)];
  } else {
    int p = tfid[row]; p = p < 0 ? 0 : (p > 9 ? 9 : p);
    v = temb[p * 32 + (c - 192)];
  }
  outp[idx] = v;
}

// ---------------------------------------------------------------------------
// Persistent word-level GRU (one workgroup per direction, h in LDS).
// 32 waves x 4 tasks cover the 128x768 gate matmul with WMMA; fused gates.
// ---------------------------------------------------------------------------
__global__ __launch_bounds__(1024)
void word_gru(const float* __restrict__ gi_f, const float* __restrict__ gi_b,
              const float* __restrict__ Whh_f, const float* __restrict__ Whh_b,
              const float* __restrict__ bhh_f, const float* __restrict__ bhh_b,
              float* __restrict__ wh, float* __restrict__ wlast)
{
  extern __shared__ float hbuf[];   // 128*256 floats
  const int dir = blockIdx.x;
  const float* gi  = dir ? gi_b  : gi_f;
  const float* Whh = dir ? Whh_b : Whh_f;
  const float* bhh = dir ? bhh_b : bhh_f;
  const int tid = threadIdx.x;
  const int wv = tid >> 5, lane = tid & 31;
  const int hf = lane >> 4, sub = lane & 15;
  for (int i = tid; i < 128 * 256; i += 1024) hbuf[i] = 0.f;
  __syncthreads();
  for (int t = 0; t < 32; ++t) {
    const int w = dir ? (31 - t) : t;
    float h2v[4][8];
    for (int task = 0; task < 4; ++task) {
      const int tile = wv * 4 + task;       // 0..127
      const int m0 = (tile >> 4) * 16;      // row tile (sequence)
      const int u0 = (tile & 15) * 16;      // hidden-unit tile
      const float* arow = hbuf + (m0 + sub) * 256 + hf * 2;
      const float* wr = Whh + (size_t)(        u0 + sub) * 256 + hf * 2;
      const float* wz = Whh + (size_t)(256   + u0 + sub) * 256 + hf * 2;
      const float* wn = Whh + (size_t)(512   + u0 + sub) * 256 + hf * 2;
      v8f ar = {}, az = {}, an = {};
#pragma unroll 2
      for (int k = 0; k < 256; k += 4) {
        v2f a  = *(const v2f*)(arow + k);
        v2f b1 = *(const v2f*)(wr + k);
        v2f b2 = *(const v2f*)(wz + k);
        v2f b3 = *(const v2f*)(wn + k);
        ar = __builtin_amdgcn_wmma_f32_16x16x4_f32(false, a, false, b1, (short)0, ar, false, false);
        az = __builtin_amdgcn_wmma_f32_16x16x4_f32(false, a, false, b2, (short)0, az, false, false);
        an = __builtin_amdgcn_wmma_f32_16x16x4_f32(false, a, false, b3, (short)0, an, false, false);
      }
      const int u = u0 + sub;
      const float br_ = bhh[u], bz_ = bhh[256 + u], bn_ = bhh[512 + u];
#pragma unroll
      for (int v = 0; v < 8; ++v) {
        const int m = m0 + v + hf * 8;
        const float* girow = gi + (size_t)(m * 32 + w) * 768;
        float r = sigmf(girow[u] + ar[v] + br_);
        float z = sigmf(girow[256 + u] + az[v] + bz_);
        float n = tanhf(girow[512 + u] + r * (an[v] + bn_));
        h2v[task][v] = (1.f - z) * n + z * hbuf[m * 256 + u];
      }
    }
    __syncthreads();
    for (int task = 0; task < 4; ++task) {
      const int tile = wv * 4 + task;
      const int m0 = (tile >> 4) * 16;
      const int u = (tile & 15) * 16 + sub;
#pragma unroll
      for (int v = 0; v < 8; ++v) {
        const int m = m0 + v + hf * 8;
        const float h2 = h2v[task][v];
        hbuf[m * 256 + u] = h2;
        wh[((size_t)(m * 32 + w)) * 512 + dir * 256 + u] = h2;
        if (t == 31) wlast[(size_t)dir * 32768 + m * 256 + u] = h2;
      }
    }
    __syncthreads();
  }
}

// ---------------------------------------------------------------------------
// Word attention -> sent_vec[128,512] (+ sentence position embedding)
// ---------------------------------------------------------------------------
__global__ __launch_bounds__(256)
void word_attn(const float* __restrict__ wh, const float* __restrict__ wa_w,
               const float* __restrict__ wa_b, const float* __restrict__ sent_pos,
               float* __restrict__ sent_vec)
{
  __shared__ float sc[32];
  const int row = blockIdx.x;             // b*16 + s
  const int tid = threadIdx.x, wv = tid >> 5, lane = tid & 31;
  for (int w = wv; w < 32; w += 8) {
    const float* r = wh + ((size_t)row * 32 + w) * 512;
    float s = 0.f;
    for (int i = lane; i < 512; i += 32) s += r[i] * wa_w[i];
    for (int off = 16; off; off >>= 1) s += __shfl_xor(s, off, 32);
    if (lane == 0) sc[w] = s + wa_b[0];
  }
  __syncthreads();
  if (tid < 32) {
    float v = sc[tid];
    float mx = v;
    for (int off = 16; off; off >>= 1) mx = fmaxf(mx, __shfl_xor(mx, off, 32));
    float e = __expf(v - mx);
    float sm = e;
    for (int off = 16; off; off >>= 1) sm += __shfl_xor(sm, off, 32);
    sc[tid] = e / sm;
  }
  __syncthreads();
  const int s_idx = row & 15;
  for (int d = tid; d < 512; d += 256) {
    float acc = 0.f;
    for (int w = 0; w < 32; ++w) acc += sc[w] * wh[((size_t)row * 32 + w) * 512 + d];
    sent_vec[(size_t)row * 512 + d] = acc + sent_pos[s_idx * 512 + d];
  }
}

// ---------------------------------------------------------------------------
// Persistent sentence-level GRU (B=8 rows padded to one 16-row WMMA tile)
// ---------------------------------------------------------------------------
__global__ __launch_bounds__(512)
void sent_gru(const float* __restrict__ gi_f, const float* __restrict__ gi_b,
              const float* __restrict__ Whh_f, const float* __restrict__ Whh_b,
              const float* __restrict__ bhh_f, const float* __restrict__ bhh_b,
              float* __restrict__ sh)
{
  __shared__ float hbuf[16 * 256];
  const int dir = blockIdx.x;
  const float* gi  = dir ? gi_b  : gi_f;
  const float* Whh = dir ? Whh_b : Whh_f;
  const float* bhh = dir ? bhh_b : bhh_f;
  const int tid = threadIdx.x, wv = tid >> 5, lane = tid & 31;
  const int hf = lane >> 4, sub = lane & 15;
  for (int i = tid; i < 16 * 256; i += 512) hbuf[i] = 0.f;
  __syncthreads();
  for (int t = 0; t < 16; ++t) {
    const int ti = dir ? (15 - t) : t;
    const int u0 = wv * 16;
    const float* arow = hbuf + sub * 256 + hf * 2;
    const float* wr = Whh + (size_t)(        u0 + sub) * 256 + hf * 2;
    const float* wz = Whh + (size_t)(256   + u0 + sub) * 256 + hf * 2;
    const float* wn = Whh + (size_t)(512   + u0 + sub) * 256 + hf * 2;
    v8f ar = {}, az = {}, an = {};
#pragma unroll 2
    for (int k = 0; k < 256; k += 4) {
      v2f a  = *(const v2f*)(arow + k);
      v2f b1 = *(const v2f*)(wr + k);
      v2f b2 = *(const v2f*)(wz + k);
      v2f b3 = *(const v2f*)(wn + k);
      ar = __builtin_amdgcn_wmma_f32_16x16x4_f32(false, a, false, b1, (short)0, ar, false, false);
      az = __builtin_amdgcn_wmma_f32_16x16x4_f32(false, a, false, b2, (short)0, az, false, false);
      an = __builtin_amdgcn_wmma_f32_16x16x4_f32(false, a, false, b3, (short)0, an, false, false);
    }
    const int u = u0 + sub;
    const float br_ = bhh[u], bz_ = bhh[256 + u], bn_ = bhh[512 + u];
    float h2v[8];
#pragma unroll
    for (int v = 0; v < 8; ++v) {
      const int m = v + hf * 8;
      if (m < 8) {
        const float* girow = gi + (size_t)(m * 16 + ti) * 768;
        float r = sigmf(girow[u] + ar[v] + br_);
        float z = sigmf(girow[256 + u] + az[v] + bz_);
        float n = tanhf(girow[512 + u] + r * (an[v] + bn_));
        h2v[v] = (1.f - z) * n + z * hbuf[m * 256 + u];
      } else h2v[v] = 0.f;
    }
    __syncthreads();
#pragma unroll
    for (int v = 0; v < 8; ++v) {
      const int m = v + hf * 8;
      if (m < 8) {
        hbuf[m * 256 + u] = h2v[v];
        sh[((size_t)(m * 16 + ti)) * 512 + dir * 256 + u] = h2v[v];
      }
    }
    __syncthreads();
  }
}

// ---------------------------------------------------------------------------
// Decoder helpers
// ---------------------------------------------------------------------------
__global__ void build_h0(const float* __restrict__ wlast, float* __restrict__ hdec)
{
  int idx = blockIdx.x * blockDim.x + threadIdx.x;
  if (idx >= 8 * 512) return;
  int b = idx >> 9, d = idx & 511;
  int seq = b * 16 + 15;
  float v = (d < 256) ? wlast[seq * 256 + d]
                      : wlast[32768 + seq * 256 + (d - 256)];
  hdec[idx] = v;
}

__global__ void dec_embed(const int* __restrict__ labels, const float* __restrict__ demb,
                          int t, float* __restrict__ xbuf, float* __restrict__ selbuf)
{
  int idx = blockIdx.x * blockDim.x + threadIdx.x;
  if (idx >= 8 * 512) return;
  int b = idx >> 9, d = idx & 511;
  int prev = (t == 0) ? 1 : labels[b * 32 + (t - 1)];
  prev = (prev >= VCAB) ? 2 : (prev < 0 ? 0 : prev);
  float o = fmaxf(demb[(size_t)prev * 512 + d], 0.f);
  xbuf[b * 1024 + d] = o;
  selbuf[b * 1536 + 512 + d] = o;
}

__global__ __launch_bounds__(256)
void attn_scores(const float* __restrict__ Wq, const float* __restrict__ Wsb,
                 const float* __restrict__ dq, const float* __restrict__ dsb,
                 const float* __restrict__ awV, const float* __restrict__ aw_vb,
                 const float* __restrict__ asV, const float* __restrict__ as_vb,
                 float* __restrict__ scw, float* __restrict__ scs)
{
  const int gw = (blockIdx.x * blockDim.x + threadIdx.x) >> 5;
  const int lane = threadIdx.x & 31;
  if (gw < 4096) {
    const int b = gw >> 9;
    const float* q = Wq + (size_t)gw * 512;
    const float* d = dq + b * 512;
    float s = 0.f;
    for (int i = lane; i < 512; i += 32) s += tanhf(q[i] + d[i]) * awV[i];
    for (int off = 16; off; off >>= 1) s += __shfl_xor(s, off, 32);
    if (lane == 0) scw[gw] = s + aw_vb[0];
  } else if (gw < 4224) {
    const int r = gw - 4096;
    const int b = r >> 4;
    const float* q = Wsb + (size_t)r * 512;
    const float* d = dsb + b * 512;
    float s = 0.f;
    for (int i = lane; i < 512; i += 32) s += tanhf(q[i] + d[i]) * asV[i];
    for (int off = 16; off; off >>= 1) s += __shfl_xor(s, off, 32);
    if (lane == 0) scs[r] = s + as_vb[0];
  }
}

__global__ __launch_bounds__(256)
void attn_combine(const float* __restrict__ scw, const float* __restrict__ scs,
                  float* __restrict__ pa)
{
  __shared__ float paw[512];
  __shared__ float pas[16];
  __shared__ float red[8];
  const int b = blockIdx.x, tid = threadIdx.x;
  if (tid < 16) {
    const int s = tid;
    const float* sw = scw + b * 512 + s * 32;
    float mx = -1e30f;
    for (int w = 0; w < 32; ++w) mx = fmaxf(mx, sw[w]);
    float sm = 0.f;
    for (int w = 0; w < 32; ++w) { float e = __expf(sw[w] - mx); paw[s * 32 + w] = e; sm += e; }
    float inv = 1.f / sm;
    for (int w = 0; w < 32; ++w) paw[s * 32 + w] *= inv;
  }
  if (tid == 31) {
    float mx = -1e30f;
    for (int s = 0; s < 16; ++s) mx = fmaxf(mx, scs[b * 16 + s]);
    float sm = 0.f;
    for (int s = 0; s < 16; ++s) { float e = __expf(scs[b * 16 + s] - mx); pas[s] = e; sm += e; }
    float inv = 1.f / sm;
    for (int s = 0; s < 16; ++s) pas[s] *= inv;
  }
  __syncthreads();
  float local = 0.f;
  for (int n = tid; n < 512; n += 256) {
    float v = paw[n] * pas[n >> 5];
    paw[n] = v;
    local += v;
  }
  for (int off = 16; off; off >>= 1) local += __shfl_xor(local, off, 32);
  if ((tid & 31) == 0) red[tid >> 5] = local;
  __syncthreads();
  if (tid == 0) {
    float tsum = 0.f;
    for (int i = 0; i < 8; ++i) tsum += red[i];
    red[0] = tsum;
  }
  __syncthreads();
  const float inv = 1.f / (red[0] + 1e-10f);
  for (int n = tid; n < 512; n += 256) pa[b * 512 + n] = paw[n] * inv;
}

__global__ __launch_bounds__(128)
void context_k(const float* __restrict__ pa, const float* __restrict__ wh,
               float* __restrict__ xbuf, float* __restrict__ selbuf)
{
  const int b = blockIdx.x;
  const int d = blockIdx.y * 128 + threadIdx.x;
  const float* p = pa + b * 512;
  const float* base = wh + ((size_t)b * 512) * 512 + d;
  float acc = 0.f;
  for (int n = 0; n < 512; ++n) acc += p[n] * base[(size_t)n * 512];
  xbuf[b * 1024 + 512 + d] = acc;
  selbuf[b * 1536 + 1024 + d] = acc;
}

__global__ void dec_gate(const float* __restrict__ gid, const float* __restrict__ ghd,
                         float* __restrict__ hdec, float* __restrict__ selbuf)
{
  int idx = blockIdx.x * blockDim.x + threadIdx.x;
  if (idx >= 8 * 512) return;
  int b = idx >> 9, u = idx & 511;
  const float* gi = gid + b * 1536;
  const float* gh = ghd + b * 1536;
  float r = sigmf(gi[u] + gh[u]);
  float z = sigmf(gi[512 + u] + gh[512 + u]);
  float n = tanhf(gi[1024 + u] + r * gh[1024 + u]);
  float h2 = (1.f - z) * n + z * hdec[idx];
  hdec[idx] = h2;
  selbuf[b * 1536 + u] = h2;
}

__global__ __launch_bounds__(256)
void pgen_k(const float* __restrict__ selp, const float* __restrict__ vs_w,
            const float* __restrict__ vs_b, float* __restrict__ pg)
{
  const int wv = threadIdx.x >> 5, lane = threadIdx.x & 31;
  if (wv < 8) {
    const float* r = selp + wv * 512;
    float s = 0.f;
    for (int i = lane; i < 512; i += 32) s += r[i] * vs_w[i];
    for (int off = 16; off; off >>= 1) s += __shfl_xor(s, off, 32);
    if (lane == 0) pg[wv] = sigmf(s + vs_b[0]);
  }
}

__global__ __launch_bounds__(1024)
void softmax_scatter(const float* __restrict__ logits, const float* __restrict__ pg,
                     const float* __restrict__ pa, const int* __restrict__ enc_ext,
                     int t, float* __restrict__ out)
{
  __shared__ float red[32];
  const int b = blockIdx.x, tid = threadIdx.x, lane = tid & 31, wv = tid >> 5;
  const float* lg = logits + (size_t)b * VCAB;
  float mx = -1e30f;
  for (int v = tid; v < VCAB; v += 1024) mx = fmaxf(mx, lg[v]);
  for (int off = 16; off; off >>= 1) mx = fmaxf(mx, __shfl_xor(mx, off, 32));
  if (lane == 0) red[wv] = mx;
  __syncthreads();
  if (tid == 0) {
    float m = red[0];
    for (int i = 1; i < 32; ++i) m = fmaxf(m, red[i]);
    red[0] = m;
  }
  __syncthreads();
  mx = red[0];
  __syncthreads();
  float sm = 0.f;
  for (int v = tid; v < VCAB; v += 1024) sm += __expf(lg[v] - mx);
  for (int off = 16; off; off >>= 1) sm += __shfl_xor(sm, off, 32);
  if (lane == 0) red[wv] = sm;
  __syncthreads();
  if (tid == 0) {
    float s = 0.f;
    for (int i = 0; i < 32; ++i) s += red[i];
    red[0] = s;
  }
  __syncthreads();
  const float inv = 1.f / red[0];
  const float pgb = pg[b];
  float* orow = out + ((size_t)b * 32 + t) * VEXTN;
  const float scale = pgb * inv;
  for (int v = tid; v < VCAB; v += 1024) orow[v] = scale * __expf(lg[v] - mx);
  for (int v = VCAB + tid; v < VEXTN; v += 1024) orow[v] = 0.f;
  __threadfence();
  __syncthreads();
  const float q = 1.f - pgb;
  for (int n = tid; n < 512; n += 1024) {
    int ix = enc_ext[b * 512 + n];
    if (ix < 0 || ix >= VEXTN) ix = 0;
    atomicAdd(orow + ix, q * pa[b * 512 + n]);
  }
}

// ---------------------------------------------------------------------------
// Host orchestration
// ---------------------------------------------------------------------------
static void gemm(const float* A, int lda, int M, const float* W, int ldw, int N,
                 int K, const float* bias, float* C, int ldc, int act, hipStream_t s)
{
  dim3 g((M + 15) / 16, (N + 15) / 16);
  gemm_f32_wmma<<<g, dim3(32), 0, s>>>(A, lda, M, W, ldw, N, K, bias, C, ldc, act);
}

extern "C" void kernel_launch(void* const* d_in, const int* in_sizes, int n_in,
                              void* d_out, int out_size, void* d_ws, size_t ws_size,
                              hipStream_t stream)
{
  (void)in_sizes; (void)n_in; (void)out_size; (void)ws_size;
  // ---- inputs ----
  const int*   input_ids = (const int*)d_in[0];
  const int*   pos_ids   = (const int*)d_in[1];
  const int*   ner_ids   = (const int*)d_in[2];
  const int*   tfidf_ids = (const int*)d_in[3];
  const int*   labels    = (const int*)d_in[4];
  const int*   enc_ext   = (const int*)d_in[6];
  // ---- params (flattened in setup_inputs insertion order) ----
  const float* word_emb  = (const float*)d_in[7];
  const float* pos_emb   = (const float*)d_in[8];
  const float* ner_emb   = (const float*)d_in[9];
  const float* tfidf_emb = (const float*)d_in[10];
  const float* wf_Wih = (const float*)d_in[11];
  const float* wf_Whh = (const float*)d_in[12];
  const float* wf_bih = (const float*)d_in[13];
  const float* wf_bhh = (const float*)d_in[14];
  const float* wb_Wih = (const float*)d_in[15];
  const float* wb_Whh = (const float*)d_in[16];
  const float* wb_bih = (const float*)d_in[17];
  const float* wb_bhh = (const float*)d_in[18];
  const float* wa_w   = (const float*)d_in[19];
  const float* wa_b   = (const float*)d_in[20];
  const float* sent_pos_emb = (const float*)d_in[21];
  const float* sf_Wih = (const float*)d_in[22];
  const float* sf_Whh = (const float*)d_in[23];
  const float* sf_bih = (const float*)d_in[24];
  const float* sf_bhh = (const float*)d_in[25];
  const float* sb_Wih = (const float*)d_in[26];
  const float* sb_Whh = (const float*)d_in[27];
  const float* sb_bih = (const float*)d_in[28];
  const float* sb_bhh = (const float*)d_in[29];
  const float* dec_emb = (const float*)d_in[30];
  const float* dg_Wih = (const float*)d_in[31];
  const float* dg_Whh = (const float*)d_in[32];
  const float* dg_bih = (const float*)d_in[33];
  const float* dg_bhh = (const float*)d_in[34];
  const float* out_w  = (const float*)d_in[35];
  const float* out_b  = (const float*)d_in[36];
  const float* aw_W1  = (const float*)d_in[37];
  const float* aw_b1  = (const float*)d_in[38];
  const float* aw_W2  = (const float*)d_in[39];
  const float* aw_b2  = (const float*)d_in[40];
  const float* aw_V   = (const float*)d_in[41];
  const float* aw_vb  = (const float*)d_in[42];
  const float* as_W1  = (const float*)d_in[43];
  const float* as_b1  = (const float*)d_in[44];
  const float* as_W2  = (const float*)d_in[45];
  const float* as_b2  = (const float*)d_in[46];
  const float* as_V   = (const float*)d_in[47];
  const float* as_vb  = (const float*)d_in[48];
  const float* sel_w  = (const float*)d_in[49];
  const float* sel_b  = (const float*)d_in[50];
  const float* vs_w   = (const float*)d_in[51];
  const float* vs_b   = (const float*)d_in[52];

  float* out = (float*)d_out;

  // ---- workspace carve-up ----
  float* ws = (float*)d_ws;
  size_t off = 0;
  auto alloc = [&](size_t n) { float* p = ws + off; off += n; return p; };
  float* emb_flat = alloc((size_t)4096 * 224);
  float* gi_wf    = alloc((size_t)4096 * 768);
  float* gi_wb    = alloc((size_t)4096 * 768);
  float* wh       = alloc((size_t)4096 * 512);   // word_hiddens [128*32, 512]
  float* wlast    = alloc((size_t)2 * 128 * 256);
  float* sent_vec = alloc((size_t)128 * 512);
  float* gi_sf    = alloc((size_t)128 * 768);
  float* gi_sb    = alloc((size_t)128 * 768);
  float* sh       = alloc((size_t)128 * 512);    // sent_hiddens [8*16, 512]
  float* Wq       = alloc((size_t)4096 * 512);
  float* Wsb      = alloc((size_t)128 * 512);
  float* hdec     = alloc((size_t)8 * 512);
  float* dq       = alloc((size_t)8 * 512);
  float* dsb      = alloc((size_t)8 * 512);
  float* scw      = alloc(4096);
  float* scs      = alloc(128);
  float* pa       = alloc((size_t)8 * 512);
  float* xbuf     = alloc((size_t)8 * 1024);
  float* gid      = alloc((size_t)8 * 1536);
  float* ghd      = alloc((size_t)8 * 1536);
  float* selbuf   = alloc((size_t)8 * 1536);
  float* selp     = alloc((size_t)8 * 512);
  float* pg       = alloc(8);
  float* logits   = alloc((size_t)8 * VCAB);

  // ---- encoder ----
  embed_concat<<<(4096 * 224 + 255) / 256, 256, 0, stream>>>(
      input_ids, pos_ids, ner_ids, tfidf_ids,
      word_emb, pos_emb, ner_emb, tfidf_emb, emb_flat);
  gemm(emb_flat, 224, 4096, wf_Wih, 224, 768, 224, wf_bih, gi_wf, 768, 0, stream);
  gemm(emb_flat, 224, 4096, wb_Wih, 224, 768, 224, wb_bih, gi_wb, 768, 0, stream);
  word_gru<<<2, 1024, 128 * 256 * sizeof(float), stream>>>(
      gi_wf, gi_wb, wf_Whh, wb_Whh, wf_bhh, wb_bhh, wh, wlast);
  word_attn<<<128, 256, 0, stream>>>(wh, wa_w, wa_b, sent_pos_emb, sent_vec);
  gemm(sent_vec, 512, 128, sf_Wih, 512, 768, 512, sf_bih, gi_sf, 768, 0, stream);
  gemm(sent_vec, 512, 128, sb_Wih, 512, 768, 512, sb_bih, gi_sb, 768, 0, stream);
  sent_gru<<<2, 512, 0, stream>>>(gi_sf, gi_sb, sf_Whh, sb_Whh, sf_bhh, sb_bhh, sh);
  // step-invariant attention projections (hoisted out of the decoder loop)
  gemm(wh, 512, 4096, aw_W1, 512, 512, 512, aw_b1, Wq, 512, 0, stream);
  gemm(sh, 512, 128, as_W1, 512, 512, 512, as_b1, Wsb, 512, 0, stream);
  build_h0<<<(8 * 512 + 255) / 256, 256, 0, stream>>>(wlast, hdec);

  // ---- decoder (teacher forcing, 32 steps) ----
  for (int t = 0; t < STN; ++t) {
    dec_embed<<<(8 * 512 + 255) / 256, 256, 0, stream>>>(labels, dec_emb, t, xbuf, selbuf);
    gemm(hdec, 512, 8, aw_W2, 512, 512, 512, aw_b2, dq, 512, 0, stream);
    gemm(hdec, 512, 8, as_W2, 512, 512, 512, as_b2, dsb, 512, 0, stream);
    attn_scores<<<528, 256, 0, stream>>>(Wq, Wsb, dq, dsb, aw_V, aw_vb, as_V, as_vb, scw, scs);
    attn_combine<<<8, 256, 0, stream>>>(scw, scs, pa);
    context_k<<<dim3(8, 4), 128, 0, stream>>>(pa, wh, xbuf, selbuf);
    gemm(xbuf, 1024, 8, dg_Wih, 1024, 1536, 1024, dg_bih, gid, 1536, 0, stream);
    gemm(hdec, 512, 8, dg_Whh, 512, 1536, 512, dg_bhh, ghd, 1536, 0, stream);
    dec_gate<<<(8 * 512 + 255) / 256, 256, 0, stream>>>(gid, ghd, hdec, selbuf);
    gemm(selbuf, 1536, 8, sel_w, 1536, 512, 1536, sel_b, selp, 512, 1, stream);
    pgen_k<<<1, 256, 0, stream>>>(selp, vs_w, vs_b, pg);
    gemm(hdec, 512, 8, out_w, 512, VCAB, 512, out_b, logits, VCAB, 0, stream);
    softmax_scatter<<<8, 1024, 0, stream>>>(logits, pg, pa, enc_ext, t, out);
  }
}